// NbModel_69801808494960
// MI455X (gfx1250) — compile-verified
//
#include <hip/hip_runtime.h>
#include <math.h>

// ---------------------------------------------------------------------------
// Reference: SIREN(32x1 line) -> L(32,16); basis separable: basis[(a,b),h,w]
//            = L[h,a]+L[w,b]; per-sample 50-step Adam LS fit; y = A.basis.
// MI455X plan: everything is latency-bound (<5MB data, ~10 GFLOP dense ->
// ~60 MFLOP after separable factorization). Kernel 1: one WG runs the 12
// sequential SIREN matmuls with V_WMMA_F32_16X16X4_F32, h kept in LDS.
// Kernel 2: 64 WGs (one per sample) run the factorized Adam loop in LDS/regs
// and emit y = P[c,h] + Q[c,w] directly.
// ---------------------------------------------------------------------------

typedef __attribute__((ext_vector_type(2))) float v2f;
typedef __attribute__((ext_vector_type(8))) float v8f;

#define DH 256
#define LSTRIDE 260   // 256 + 4 pad floats: 8B-aligned rows, conflict-free banks

__device__ __forceinline__ v8f wmma_f32_k4(v2f a, v2f b, v8f c) {
  // D = A(16x4 f32) * B(4x16 f32) + C(16x16 f32)
  return __builtin_amdgcn_wmma_f32_16x16x4_f32(
      /*neg_a=*/false, a, /*neg_b=*/false, b,
      /*c_mod=*/(short)0, c, /*reuse_a=*/false, /*reuse_b=*/false);
}

// ---------------------------------------------------------------------------
// Kernel 1: SIREN -> L (32 x 16) into workspace.  Single workgroup, 8 waves.
// out[m,n] = sum_k h[m,k] * W[n,k]  (i.e. B[k,n] = W[n,k], contiguous in k).
// A frag (16x4 f32, ISA 7.12.2): lane l<16 -> {h[m0+l,k], h[m0+l,k+1]},
//   lane l+16 -> {h[m0+l,k+2], h[m0+l,k+3]}  == one float2 load per lane.
// B frag mirrored: lane l<16 -> {W[n0+l,k], W[n0+l,k+1]}, lane l+16 -> k+2,k+3.
// C/D: row = m0 + 8*(lane>=16) + vgpr_j, col = n0 + (lane&15).
// ---------------------------------------------------------------------------
__global__ __launch_bounds__(256)
void siren_kernel(const float* __restrict__ W0, const float* __restrict__ b0,
                  const float* __restrict__ Wh, const float* __restrict__ bh,
                  const float* __restrict__ Wl, const float* __restrict__ bl,
                  float* __restrict__ Lout) {
  __shared__ float sh[32 * LSTRIDE];   // h (32 x 256), padded

  const int tid  = threadIdx.x;
  const int wave = tid >> 5;
  const int lane = tid & 31;
  const int lr   = lane & 15;
  const int half = lane >> 4;

  // First layer (K=1, elementwise): h[m,n] = sin(30*(line[m]*W0[n] + b0[n]))
  for (int idx = tid; idx < 32 * DH; idx += 256) {
    int m = idx >> 8, n = idx & 255;
    float line = -1.0f + 2.0f * (float)m / 31.0f;
    sh[m * LSTRIDE + n] = sinf(30.0f * (line * W0[n] + b0[n]));
  }
  __syncthreads();

  v8f vzero;
#pragma unroll
  for (int j = 0; j < 8; ++j) vzero[j] = 0.0f;

  // 11 hidden layers: h = sin(h @ Wh[l]^T + bh[l]); in-place via reg staging.
  for (int l = 0; l < 11; ++l) {
    const float* W    = Wh + l * DH * DH;
    const float* bias = bh + l * DH;

    v8f acc[4];
#pragma unroll
    for (int ti = 0; ti < 4; ++ti) {
      acc[ti] = vzero;
      int tile  = wave * 4 + ti;      // 8 waves x 4 tiles = 2 (M) x 16 (N)
      int mbase = (tile & 1) * 16;
      int nbase = (tile >> 1) * 16;
      const float* hrow = &sh[(mbase + lr) * LSTRIDE + 2 * half];
      const float* wrow = &W[(nbase + lr) * DH + 2 * half];
      for (int k = 0; k < DH; k += 4) {
        v2f a = *(const v2f*)(hrow + k);
        v2f b = *(const v2f*)(wrow + k);
        acc[ti] = wmma_f32_k4(a, b, acc[ti]);
      }
    }
    __syncthreads();                  // all reads of sh complete
#pragma unroll
    for (int ti = 0; ti < 4; ++ti) {
      int tile  = wave * 4 + ti;
      int mbase = (tile & 1) * 16;
      int nbase = (tile >> 1) * 16;
      int col   = nbase + lr;
      float bv  = bias[col];
#pragma unroll
      for (int j = 0; j < 8; ++j) {
        int row = mbase + half * 8 + j;
        sh[row * LSTRIDE + col] = sinf(acc[ti][j] + bv);
      }
    }
    __syncthreads();                  // sh fully updated for next layer
  }

  // Head: L = h @ Wl^T + bl  -> (32 x 16). Waves 0,1 each do one 16x16 tile.
  if (wave < 2) {
    int mbase = wave * 16;
    v8f acc = vzero;
    const float* hrow = &sh[(mbase + lr) * LSTRIDE + 2 * half];
    const float* wrow = &Wl[lr * DH + 2 * half];
    for (int k = 0; k < DH; k += 4) {
      v2f a = *(const v2f*)(hrow + k);
      v2f b = *(const v2f*)(wrow + k);
      acc = wmma_f32_k4(a, b, acc);
    }
    float bv = bl[lr];
#pragma unroll
    for (int j = 0; j < 8; ++j) {
      int row = mbase + half * 8 + j;
      Lout[row * 16 + lr] = acc[j] + bv;   // L[h][a]
    }
  }
}

// ---------------------------------------------------------------------------
// Kernel 2: per-sample separable Adam fit + reconstruction.
//   y[c,h,w]      = P[c,h] + Q[c,w],  P = Arow@L^T, Q = Acol@L^T
//   Rh[c,h]       = 32*P + sumQ[c] - Xh[c,h]   (Xh = sum_w x)
//   g[(a,b),c]    = (2/denom)*(Gh[c,a] + Gw[c,b]),  Gh = Rh@L, Gw = Rw@L
// Adam moments live in registers (3 per thread), A in LDS (one k per thread).
// ---------------------------------------------------------------------------
#define AB1   0.9f
#define AB2   0.999f
#define ALR   0.1f
#define AEPS  1e-8f
#define GSCALE (2.0f / 196608.0f)     // 2 / x.size, x.size = 64*3*32*32

__global__ __launch_bounds__(256)
void adam_kernel(const float* __restrict__ x, const float* __restrict__ Lg,
                 float* __restrict__ out) {
  __shared__ float sA[256 * 3];       // A[k][c]
  __shared__ float sL[32 * 16];       // L[h][a]
  __shared__ float Xh[96], Xw[96];    // [c*32 + h/w]
  __shared__ float Arow[48], Acol[48];
  __shared__ float P[96], Q[96], Rh[96], Rw[96];
  __shared__ float Gh[48], Gw[48];
  __shared__ float sums[6];           // [0..2]=sumP, [3..5]=sumQ

  const int tid = threadIdx.x;
  const int s   = blockIdx.x;
  const float* xs = x + s * 3 * 32 * 32;

  for (int i = tid; i < 512; i += 256) sL[i] = Lg[i];
#pragma unroll
  for (int c = 0; c < 3; ++c) sA[tid * 3 + c] = 1.0f / 256.0f;

  if (tid < 96) {                               // Xh[c,h] = sum_w x
    int c = tid / 32, h = tid % 32;
    float a = 0.0f;
    for (int w = 0; w < 32; ++w) a += xs[(c * 32 + h) * 32 + w];
    Xh[tid] = a;
  } else if (tid >= 128 && tid < 224) {         // Xw[c,w] = sum_h x
    int u = tid - 128, c = u / 32, w = u % 32;
    float a = 0.0f;
    for (int h = 0; h < 32; ++h) a += xs[(c * 32 + h) * 32 + w];
    Xw[u] = a;
  }
  __syncthreads();

  float m[3] = {0.0f, 0.0f, 0.0f};
  float v[3] = {0.0f, 0.0f, 0.0f};
  float b1t = 1.0f, b2t = 1.0f;
  const int ka = tid >> 4, kb = tid & 15;       // k = a*16 + b

  for (int t = 1; t <= 50; ++t) {
    b1t *= AB1; b2t *= AB2;

    if (tid < 48) {                             // Arow[c,a] = sum_b A
      int c = tid >> 4, a = tid & 15;
      float acc = 0.0f;
      for (int b = 0; b < 16; ++b) acc += sA[((a << 4) + b) * 3 + c];
      Arow[tid] = acc;
    } else if (tid >= 64 && tid < 112) {        // Acol[c,b] = sum_a A
      int u = tid - 64, c = u >> 4, b = u & 15;
      float acc = 0.0f;
      for (int a = 0; a < 16; ++a) acc += sA[((a << 4) + b) * 3 + c];
      Acol[u] = acc;
    }
    __syncthreads();

    if (tid < 96) {                             // P[c,h] = sum_a Arow*L
      int c = tid / 32, h = tid % 32;
      float p = 0.0f;
      for (int a = 0; a < 16; ++a) p += Arow[c * 16 + a] * sL[h * 16 + a];
      P[tid] = p;
    } else if (tid >= 128 && tid < 224) {       // Q[c,w] = sum_b Acol*L
      int u = tid - 128, c = u / 32, w = u % 32;
      float q = 0.0f;
      for (int b = 0; b < 16; ++b) q += Acol[c * 16 + b] * sL[w * 16 + b];
      Q[u] = q;
    }
    __syncthreads();

    if (tid < 3) {
      float a = 0.0f;
      for (int h = 0; h < 32; ++h) a += P[tid * 32 + h];
      sums[tid] = a;
    } else if (tid >= 4 && tid < 7) {
      int c = tid - 4;
      float a = 0.0f;
      for (int w = 0; w < 32; ++w) a += Q[c * 32 + w];
      sums[3 + c] = a;
    }
    __syncthreads();

    if (tid < 96) {                             // residual row/col sums
      int c = tid / 32;
      Rh[tid] = 32.0f * P[tid] + sums[3 + c] - Xh[tid];
    } else if (tid >= 128 && tid < 224) {
      int u = tid - 128, c = u / 32;
      Rw[u] = 32.0f * Q[u] + sums[c] - Xw[u];
    }
    __syncthreads();

    if (tid < 48) {                             // Gh[c,a] = sum_h Rh*L
      int c = tid >> 4, a = tid & 15;
      float g = 0.0f;
      for (int h = 0; h < 32; ++h) g += Rh[c * 32 + h] * sL[h * 16 + a];
      Gh[tid] = g;
    } else if (tid >= 64 && tid < 112) {        // Gw[c,b] = sum_w Rw*L
      int u = tid - 64, c = u >> 4, b = u & 15;
      float g = 0.0f;
      for (int w = 0; w < 32; ++w) g += Rw[c * 32 + w] * sL[w * 16 + b];
      Gw[u] = g;
    }
    __syncthreads();

#pragma unroll
    for (int c = 0; c < 3; ++c) {               // Adam update, k = tid
      float g = GSCALE * (Gh[c * 16 + ka] + Gw[c * 16 + kb]);
      m[c] = AB1 * m[c] + (1.0f - AB1) * g;
      v[c] = AB2 * v[c] + (1.0f - AB2) * g * g;
      float mh = m[c] / (1.0f - b1t);
      float vh = v[c] / (1.0f - b2t);
      sA[tid * 3 + c] -= ALR * mh / (sqrtf(vh) + AEPS);
    }
    __syncthreads();
  }

  // Final P,Q from converged A, then y[c,h,w] = P[c,h] + Q[c,w]
  if (tid < 48) {
    int c = tid >> 4, a = tid & 15;
    float acc = 0.0f;
    for (int b = 0; b < 16; ++b) acc += sA[((a << 4) + b) * 3 + c];
    Arow[tid] = acc;
  } else if (tid >= 64 && tid < 112) {
    int u = tid - 64, c = u >> 4, b = u & 15;
    float acc = 0.0f;
    for (int a = 0; a < 16; ++a) acc += sA[((a << 4) + b) * 3 + c];
    Acol[u] = acc;
  }
  __syncthreads();
  if (tid < 96) {
    int c = tid / 32, h = tid % 32;
    float p = 0.0f;
    for (int a = 0; a < 16; ++a) p += Arow[c * 16 + a] * sL[h * 16 + a];
    P[tid] = p;
  } else if (tid >= 128 && tid < 224) {
    int u = tid - 128, c = u / 32, w = u % 32;
    float q = 0.0f;
    for (int b = 0; b < 16; ++b) q += Acol[c * 16 + b] * sL[w * 16 + b];
    Q[u] = q;
  }
  __syncthreads();

  float* ys = out + s * 3072;
  for (int idx = tid; idx < 3072; idx += 256) {
    int c = idx >> 10, r = idx & 1023, h = r >> 5, w = r & 31;
    ys[idx] = P[c * 32 + h] + Q[c * 32 + w];
  }
}

// ---------------------------------------------------------------------------
extern "C" void kernel_launch(void* const* d_in, const int* in_sizes, int n_in,
                              void* d_out, int out_size, void* d_ws, size_t ws_size,
                              hipStream_t stream) {
  (void)in_sizes; (void)n_in; (void)out_size; (void)ws_size;
  const float* x  = (const float*)d_in[0];   // (64,3,32,32)
  const float* W0 = (const float*)d_in[1];   // (256,1)
  const float* b0 = (const float*)d_in[2];   // (256,)
  const float* Wh = (const float*)d_in[3];   // (11,256,256)
  const float* bh = (const float*)d_in[4];   // (11,256)
  const float* Wl = (const float*)d_in[5];   // (16,256)
  const float* bl = (const float*)d_in[6];   // (16,)
  float* out = (float*)d_out;                // (64,3,32,32)
  float* Lws = (float*)d_ws;                 // L (32,16) scratch

  siren_kernel<<<1, 256, 0, stream>>>(W0, b0, Wh, bh, Wl, bl, Lws);
  adam_kernel<<<64, 256, 0, stream>>>(x, Lws, out);
}